// GCNLayer_69372311765406
// MI455X (gfx1250) — compile-verified
//
#include <hip/hip_runtime.h>

#define NN 8192
#define EE 262144
#define DD 128
#define WORDS (NN / 32) /* 256 u32 per bitmap row */

typedef float v2f __attribute__((ext_vector_type(2)));
typedef float v8f __attribute__((ext_vector_type(8)));

// ---------------------------------------------------------------------------
// 1) Zero the adjacency bitmap (8 MB of workspace), vectorized stores.
// ---------------------------------------------------------------------------
__global__ void gcn_zero_bits(unsigned int* __restrict__ bits) {
    size_t i = (size_t)blockIdx.x * blockDim.x + threadIdx.x;
    size_t nvec = (size_t)NN * WORDS / 4;
    uint4* p = (uint4*)bits;
    if (i < nvec) p[i] = make_uint4(0u, 0u, 0u, 0u);
}

// ---------------------------------------------------------------------------
// 2) Scatter edges + self loops into the bitmap. atomicOr is idempotent and
//    commutative -> deterministic result and exact deduplication, matching
//    the reference's scatter-set semantics.
// ---------------------------------------------------------------------------
__global__ void gcn_scatter_edges(const long long* __restrict__ ei,
                                  unsigned int* __restrict__ bits) {
    int t = blockIdx.x * blockDim.x + threadIdx.x;
    if (t < EE) {
        int src = (int)ei[t];
        int dst = (int)ei[EE + t];
        atomicOr(&bits[(size_t)src * WORDS + (dst >> 5)], 1u << (dst & 31));
    } else {
        int i = t - EE;
        if (i < NN) atomicOr(&bits[(size_t)i * WORDS + (i >> 5)], 1u << (i & 31));
    }
}

// ---------------------------------------------------------------------------
// 3) Degree (popcount of deduped row) -> d^-1/2. deg > 0 guaranteed by the
//    self loop, so the where() branch of the reference is always the rsqrt.
// ---------------------------------------------------------------------------
__global__ void gcn_degrees(const unsigned int* __restrict__ bits,
                            float* __restrict__ dinv) {
    int i = blockIdx.x * blockDim.x + threadIdx.x;
    if (i >= NN) return;
    const uint4* row = (const uint4*)(bits + (size_t)i * WORDS);
    int d = 0;
#pragma unroll 8
    for (int w = 0; w < WORDS / 4; ++w) {
        uint4 v = row[w];
        d += __popc(v.x) + __popc(v.y) + __popc(v.z) + __popc(v.w);
    }
    dinv[i] = 1.0f / sqrtf((float)d);
}

// ---------------------------------------------------------------------------
// 4) support = x @ W via V_WMMA_F32_16X16X4_F32 (f32 matrix pipe, faithful
//    to the f32 reference). Block = 256 threads = 8 waves; block b owns the
//    16-row stripe m_tile = b and its 8 column tiles (one per wave).
//    W (64 KB) is staged once in LDS per block.
//
//    A-frag (16x4 f32, ISA 7.12.2): lanes 0-15 -> M rows, K=0,1 in v[0],v[1];
//                                   lanes 16-31 -> same M rows, K=2,3.
//    B-frag (4x16 f32) mirrored;  C/D: 8 VGPRs, lanes 0-15 M=r, 16-31 M=r+8.
// ---------------------------------------------------------------------------
__global__ __launch_bounds__(256) void gcn_support_wmma(
    const float* __restrict__ x, const float* __restrict__ w,
    float* __restrict__ support) {
    __shared__ float sW[DD * DD];

    int tid = threadIdx.x;
    {
        float4* s4 = (float4*)sW;
        const float4* w4 = (const float4*)w;
        for (int i = tid; i < DD * DD / 4; i += 256) s4[i] = w4[i];
    }
    __syncthreads();

    int wave = tid >> 5;
    int lane = tid & 31;
    int tile = blockIdx.x * 8 + wave;
    int m_tile = tile >> 3; // 0..511
    int n_tile = tile & 7;  // 0..7
    int half = lane >> 4;   // K-pair select
    int l16 = lane & 15;
    int row = m_tile * 16 + l16;
    int col = n_tile * 16 + l16;

    v8f c = {};
#pragma unroll 4
    for (int k0 = 0; k0 < DD; k0 += 4) {
        int ka = k0 + half * 2;
        v2f a = {x[row * DD + ka], x[row * DD + ka + 1]};
        v2f b = {sW[ka * DD + col], sW[(ka + 1) * DD + col]};
        // 8 args: (neg_a, A, neg_b, B, c_mod, C, reuse_a, reuse_b)
        c = __builtin_amdgcn_wmma_f32_16x16x4_f32(false, a, false, b,
                                                  (short)0, c, false, false);
    }

    int mbase = m_tile * 16 + half * 8;
#pragma unroll
    for (int r = 0; r < 8; ++r) {
        support[(mbase + r) * DD + col] = c[r];
    }
}

// ---------------------------------------------------------------------------
// 5) out[i] = dinv[i] * sum_{j in adj(i)} dinv[j] * support[j] + bias.
//    One 128-thread block per node, lane = channel. In-order bit scan over
//    the bitmap row -> ascending-j accumulation, fully deterministic.
//    support (4 MB) is L2-resident on MI455X (192 MB L2), so the ~33
//    gathered rows per node are cheap.
// ---------------------------------------------------------------------------
__global__ __launch_bounds__(128) void gcn_aggregate(
    const unsigned int* __restrict__ bits, const float* __restrict__ dinv,
    const float* __restrict__ support, const float* __restrict__ bias,
    float* __restrict__ out) {
    int i = blockIdx.x;  // node
    int t = threadIdx.x; // channel
    const unsigned int* row = bits + (size_t)i * WORDS;

    float acc = 0.0f;
    for (int w = 0; w < WORDS; ++w) {
        unsigned int word = row[w]; // broadcast load, one fetch per wave
        while (word) {
            int b = __builtin_ctz(word);
            word &= word - 1;
            int j = (w << 5) + b;
            acc += dinv[j] * support[j * DD + t];
        }
    }
    out[i * DD + t] = dinv[i] * acc + bias[t];
}

// ---------------------------------------------------------------------------
extern "C" void kernel_launch(void* const* d_in, const int* in_sizes, int n_in,
                              void* d_out, int out_size, void* d_ws,
                              size_t ws_size, hipStream_t stream) {
    (void)in_sizes; (void)n_in; (void)out_size; (void)ws_size;

    const float* x = (const float*)d_in[0];          // [N, DIN] f32
    const float* w = (const float*)d_in[1];          // [DIN, DOUT] f32
    const float* bias = (const float*)d_in[2];       // [DOUT] f32
    const long long* ei = (const long long*)d_in[3]; // [2, E] int64
    float* out = (float*)d_out;                      // [N, DOUT] f32

    // Workspace layout: bitmap (8 MB) | dinv (32 KB) | support (4 MB)
    unsigned char* ws = (unsigned char*)d_ws;
    unsigned int* bits = (unsigned int*)ws;
    size_t off = (size_t)NN * WORDS * sizeof(unsigned int);
    float* dinv = (float*)(ws + off);
    off += (size_t)NN * sizeof(float);
    float* support = (float*)(ws + off);

    {
        int nvec = NN * WORDS / 4;
        hipLaunchKernelGGL(gcn_zero_bits, dim3((nvec + 255) / 256), dim3(256),
                           0, stream, bits);
    }
    hipLaunchKernelGGL(gcn_scatter_edges, dim3((EE + NN + 255) / 256),
                       dim3(256), 0, stream, ei, bits);
    hipLaunchKernelGGL(gcn_degrees, dim3(NN / 256), dim3(256), 0, stream, bits,
                       dinv);
    hipLaunchKernelGGL(gcn_support_wmma, dim3((NN / 16) * (DD / 16) / 8),
                       dim3(256), 0, stream, x, w, support);
    hipLaunchKernelGGL(gcn_aggregate, dim3(NN), dim3(128), 0, stream, bits,
                       dinv, support, bias, out);
}